// spGAT_17171279249892
// MI455X (gfx1250) — compile-verified
//
#include <hip/hip_runtime.h>
#include <math.h>

#define NN 100000
#define EE 1600000
#define GG 1000
#define DH 128
#define LRELU_SLOPE 0.2f
#define BN_EPS_C 1e-5f

typedef float v2f __attribute__((ext_vector_type(2)));
typedef float v8f __attribute__((ext_vector_type(8)));

// ---------------------------------------------------------------------------
// C[n, 0:128] = A[n, 0:128] * B[128, 128] (+bias). Row-major A, B, C.
// Block = 8 waves -> 16 rows x full 128 cols; wave w owns 16-col tile w.
// A tile (16x128) is staged into LDS with gfx1250 async load-to-LDS, then
// each wave runs 32 v_wmma_f32_16x16x4_f32 K-steps.
// A 16x4 layout : lanes 0-15 rows M=lane (K=k0,k0+1), lanes 16-31 K=k0+2,k0+3.
// B 4x16 layout : symmetric; C/D: VGPR r -> (M=r | r+8, N=lane&15).
// Requires gridDim.x * 16 == nrows (exact tiling; NN % 16 == 0).
// ---------------------------------------------------------------------------
__global__ __launch_bounds__(256) void gemm_n128_wmma(
    const float* __restrict__ A, const float* __restrict__ B,
    const float* __restrict__ bias, float* __restrict__ C)
{
  __shared__ float Ash[16 * DH];           // 8 KB

  const int rowBase = blockIdx.x << 4;

  // ---- async-stage the 16x128 fp32 A tile into LDS (2 x B128 per thread) ----
  {
    const int t = threadIdx.x;
    #pragma unroll
    for (int i = 0; i < 2; ++i) {
      const int idx4 = t + (i << 8);       // float4 index 0..511
      const int row  = idx4 >> 5;          // 32 float4 per 128-wide row
      const int c4   = idx4 & 31;
      const float* gp = A + (size_t)(rowBase + row) * DH + (c4 << 2);
      const unsigned ldsoff = (unsigned)(size_t)(&Ash[idx4 << 2]);
      asm volatile("global_load_async_to_lds_b128 %0, %1, off"
                   :: "v"(ldsoff), "v"((unsigned long long)(size_t)gp)
                   : "memory");
    }
    asm volatile("s_wait_asynccnt 0x0" ::: "memory");
  }
  __syncthreads();

  const int wave = threadIdx.x >> 5;
  const int lane = threadIdx.x & 31;
  const int colBase = wave << 4;
  const int m    = lane & 15;
  const int half = lane >> 4;              // 0: K pair 0/1, 1: K pair 2/3
  const int kb   = half << 1;

  const float bv = bias ? bias[colBase + m] : 0.0f;
  v8f acc;
  #pragma unroll
  for (int r = 0; r < 8; ++r) acc[r] = bv;

  const float* __restrict__ As = &Ash[m * DH];
  #pragma unroll 4
  for (int k0 = 0; k0 < DH; k0 += 4) {
    v2f a, b;
    a.x = As[k0 + kb];
    a.y = As[k0 + kb + 1];
    b.x = B[(size_t)(k0 + kb)     * DH + colBase + m];
    b.y = B[(size_t)(k0 + kb + 1) * DH + colBase + m];
    acc = __builtin_amdgcn_wmma_f32_16x16x4_f32(
        /*neg_a=*/false, a, /*neg_b=*/false, b,
        /*c_mod=*/(short)0, acc, /*reuse_a=*/false, /*reuse_b=*/false);
  }

  const int col = colBase + m;
  float* __restrict__ Cp = C + (size_t)(rowBase + (half << 3)) * DH + col;
  #pragma unroll
  for (int r = 0; r < 8; ++r) Cp[(size_t)r * DH] = acc[r];
}

// ---------------------------------------------------------------------------
__global__ void fill_f32(float* __restrict__ p, long n, float v) {
  long i = (long)blockIdx.x * blockDim.x + threadIdx.x;
  if (i < n) p[i] = v;
}

// Permute W[h][i][o] ([8,128,16]) -> Bm[i][h*16+o] ([128,128])
__global__ void pack_w8(const float* __restrict__ W, float* __restrict__ Bm) {
  int t = blockIdx.x * blockDim.x + threadIdx.x;
  if (t >= DH * DH) return;
  int i = t >> 7, col = t & 127, hh = col >> 4, o = col & 15;
  Bm[t] = W[((hh << 7) + i) * 16 + o];
}

// es[n,h] = dot(Wh[n, h*dlen : (h+1)*dlen], a_src[h]); same for ed.
template <int H, int LH>
__global__ void attn_proj(const float* __restrict__ Wh,
                          const float* __restrict__ a_s, const float* __restrict__ a_d,
                          float* __restrict__ es, float* __restrict__ ed) {
  int t = blockIdx.x * blockDim.x + threadIdx.x;
  if (t >= NN * H) return;
  int n = t >> LH;
  int h = t & (H - 1);
  const int dlen = DH / H;
  const float* w  = Wh  + (size_t)n * DH + h * dlen;
  const float* as = a_s + h * dlen;
  const float* ad = a_d + h * dlen;
  float ss = 0.f, sd = 0.f;
  #pragma unroll
  for (int o = 0; o < dlen; ++o) { float x = w[o]; ss += x * as[o]; sd += x * ad[o]; }
  es[t] = ss; ed[t] = sd;
}

__device__ __forceinline__ float leakyr(float v) { return v > 0.f ? v : LRELU_SLOPE * v; }

// float atomic max via sign-aware int atomics (valid with -inf init)
__device__ __forceinline__ void atomicMaxF(float* addr, float v) {
  if (v >= 0.f) atomicMax((int*)addr, __float_as_int(v));
  else atomicMin((unsigned int*)addr, (unsigned int)__float_as_int(v));
}

template <int H>
__global__ void edge_max_k(const int* __restrict__ src, const int* __restrict__ dst,
                           const float* __restrict__ es, const float* __restrict__ ed,
                           float* __restrict__ emax) {
  int e = blockIdx.x * blockDim.x + threadIdx.x;
  if (e >= EE) return;
  int s = src[e], d = dst[e];
  #pragma unroll
  for (int h = 0; h < H; ++h) {
    float v = leakyr(es[s * H + h] + ed[d * H + h]);
    atomicMaxF(&emax[d * H + h], v);
  }
}

template <int H>
__global__ void edge_den_k(const int* __restrict__ src, const int* __restrict__ dst,
                           const float* __restrict__ es, const float* __restrict__ ed,
                           const float* __restrict__ emax, float* __restrict__ denom) {
  int e = blockIdx.x * blockDim.x + threadIdx.x;
  if (e >= EE) return;
  int s = src[e], d = dst[e];
  #pragma unroll
  for (int h = 0; h < H; ++h) {
    float v = leakyr(es[s * H + h] + ed[d * H + h]);
    atomicAdd(&denom[d * H + h], expf(v - emax[d * H + h]));
  }
}

// 32 lanes per edge; each lane handles 4 contiguous channels (one head group).
template <int H, int SHIFT>
__global__ void edge_agg_k(const int* __restrict__ src, const int* __restrict__ dst,
                           const float* __restrict__ es, const float* __restrict__ ed,
                           const float* __restrict__ emax, const float* __restrict__ denom,
                           const float* __restrict__ Wh, float* __restrict__ agg) {
  long t = (long)blockIdx.x * blockDim.x + threadIdx.x;
  if (t >= (long)EE * 32) return;
  int e = (int)(t >> 5);
  int c = ((int)t & 31) << 2;
  int s = src[e], d = dst[e];
  int h = c >> SHIFT;
  float v   = leakyr(es[s * H + h] + ed[d * H + h]);
  float ee  = expf(v - emax[d * H + h]);
  float att = ee / (denom[d * H + h] + 1e-9f);
  const float4 wv = *(const float4*)(Wh + (size_t)s * DH + c);
  float* ap = agg + (size_t)d * DH + c;
  atomicAdd(ap + 0, wv.x * att);
  atomicAdd(ap + 1, wv.y * att);
  atomicAdd(ap + 2, wv.z * att);
  atomicAdd(ap + 3, wv.w * att);
}

// Per-feature sum / sumsq of (agg * n_norm) over all nodes -> acc[0:128], acc[128:256]
__global__ void bn_stats(const float* __restrict__ agg, const float* __restrict__ n_norm,
                         float* __restrict__ acc) {
  __shared__ float s1[256], s2[256];
  int c  = threadIdx.x & 127;
  int ln = threadIdx.x >> 7;
  float a = 0.f, b = 0.f;
  for (int n = blockIdx.x * 2 + ln; n < NN; n += gridDim.x * 2) {
    float v = agg[(size_t)n * DH + c] * n_norm[n];
    a += v; b += v * v;
  }
  s1[threadIdx.x] = a; s2[threadIdx.x] = b;
  __syncthreads();
  if (threadIdx.x < 128) {
    atomicAdd(&acc[c],       s1[threadIdx.x] + s1[threadIdx.x + 128]);
    atomicAdd(&acc[128 + c], s2[threadIdx.x] + s2[threadIdx.x + 128]);
  }
}

// h = h + elu(gamma*(agg*n_norm - mu)*rsqrt(var+eps) + beta)
__global__ void bn_apply(float* __restrict__ h, const float* __restrict__ agg,
                         const float* __restrict__ n_norm,
                         const float* __restrict__ gamma, const float* __restrict__ beta,
                         const float* __restrict__ acc) {
  long t = (long)blockIdx.x * blockDim.x + threadIdx.x;
  if (t >= (long)NN * DH) return;
  int n = (int)(t >> 7), c = (int)(t & 127);
  float v   = agg[t] * n_norm[n];
  float mu  = acc[c] * (1.0f / NN);
  float var = acc[128 + c] * (1.0f / NN) - mu * mu;
  float xn  = gamma[c] * (v - mu) * rsqrtf(var + BN_EPS_C) + beta[c];
  float e   = xn > 0.f ? xn : (expf(xn) - 1.f);
  h[t] = h[t] + e;
}

__global__ void pool_k(const float* __restrict__ h, const int* __restrict__ indicator,
                       float* __restrict__ pool, float* __restrict__ cnt) {
  long t = (long)blockIdx.x * blockDim.x + threadIdx.x;
  if (t >= (long)NN * DH) return;
  int n = (int)(t >> 7), c = (int)(t & 127);
  int g = indicator[n];
  atomicAdd(&pool[(size_t)g * DH + c], h[t]);
  if (c == 0) atomicAdd(&cnt[g], 1.0f);
}

__global__ void pool_div(float* __restrict__ pool, const float* __restrict__ cnt) {
  int t = blockIdx.x * blockDim.x + threadIdx.x;
  if (t >= GG * DH) return;
  pool[t] /= fmaxf(cnt[t >> 7], 1.0f);
}

__global__ void mlp_layer(const float* __restrict__ in, const float* __restrict__ W,
                          const float* __restrict__ b, float* __restrict__ out,
                          int K, int M, int relu) {
  int t = blockIdx.x * blockDim.x + threadIdx.x;
  if (t >= GG * M) return;
  int g = t / M, j = t % M;
  const float* row = in + (size_t)g * K;
  float s = b[j];
  for (int i = 0; i < K; ++i) s += row[i] * W[i * M + j];
  out[t] = relu ? fmaxf(s, 0.f) : s;
}

// ---------------------------------------------------------------------------
extern "C" void kernel_launch(void* const* d_in, const int* in_sizes, int n_in,
                              void* d_out, int out_size, void* d_ws, size_t ws_size,
                              hipStream_t stream) {
  (void)in_sizes; (void)n_in; (void)out_size; (void)ws_size;

  const float* x         = (const float*)d_in[0];
  const int*   edges     = (const int*)  d_in[1];
  const int*   src       = edges;
  const int*   dst       = edges + EE;
  const int*   indicator = (const int*)  d_in[2];
  const float* n_norm    = (const float*)d_in[3];
  const float* embed_W   = (const float*)d_in[5];
  const float* embed_b   = (const float*)d_in[6];
  const float* W03       = (const float*)d_in[7];
  const float* a_src03   = (const float*)d_in[8];
  const float* a_dst03   = (const float*)d_in[9];
  const float* gamma03   = (const float*)d_in[10];
  const float* beta03    = (const float*)d_in[11];
  const float* W3        = (const float*)d_in[12];
  const float* a_src3    = (const float*)d_in[13];
  const float* a_dst3    = (const float*)d_in[14];
  const float* gamma3    = (const float*)d_in[15];
  const float* beta3     = (const float*)d_in[16];
  const float* mlp_W0    = (const float*)d_in[17];
  const float* mlp_b0    = (const float*)d_in[18];
  const float* mlp_W1    = (const float*)d_in[19];
  const float* mlp_b1    = (const float*)d_in[20];
  const float* mlp_W2    = (const float*)d_in[21];
  const float* mlp_b2    = (const float*)d_in[22];
  float* out = (float*)d_out;

  size_t off = 0;
  auto alloc = [&](size_t bytes) -> float* {
    char* p = (char*)d_ws + off;
    off += (bytes + 255) & ~(size_t)255;
    return (float*)p;
  };
  float* h     = alloc((size_t)NN * DH * 4);
  float* Wh    = alloc((size_t)NN * DH * 4);
  float* agg   = alloc((size_t)NN * DH * 4);
  float* es    = alloc((size_t)NN * 8 * 4);
  float* ed    = alloc((size_t)NN * 8 * 4);
  float* emax  = alloc((size_t)NN * 8 * 4);
  float* denom = alloc((size_t)NN * 8 * 4);
  float* Bm    = alloc((size_t)DH * DH * 4);
  float* acc   = alloc(256 * 4);
  float* pool  = alloc((size_t)GG * DH * 4);
  float* cnt   = alloc((size_t)GG * 4);
  float* t0    = alloc((size_t)GG * 64 * 4);
  float* t1    = alloc((size_t)GG * 32 * 4);

  const int TB = 256;
  const int gGemm  = NN / 16;                              // 6250 (exact)
  const int gE     = (EE + TB - 1) / TB;
  const long nAgg  = (long)EE * 32;
  const int gAgg   = (int)((nAgg + TB - 1) / TB);          // 200000
  const long nNC   = (long)NN * DH;
  const int gNC    = (int)((nNC + TB - 1) / TB);

  // 1) embedding: h = x @ embed_W + embed_b
  gemm_n128_wmma<<<gGemm, TB, 0, stream>>>(x, embed_W, embed_b, h);

  auto gat_common = [&](const float* gm, const float* bt) {
    fill_f32<<<1, TB, 0, stream>>>(acc, 256, 0.f);
    bn_stats<<<512, TB, 0, stream>>>(agg, n_norm, acc);
    bn_apply<<<gNC, TB, 0, stream>>>(h, agg, n_norm, gm, bt, acc);
  };

  // 2) three 8-head GAT layers
  for (int l = 0; l < 3; ++l) {
    pack_w8<<<(DH * DH + TB - 1) / TB, TB, 0, stream>>>(W03 + (size_t)l * 8 * DH * 16, Bm);
    gemm_n128_wmma<<<gGemm, TB, 0, stream>>>(h, Bm, nullptr, Wh);
    long nH = (long)NN * 8;
    attn_proj<8, 3><<<(int)((nH + TB - 1) / TB), TB, 0, stream>>>(
        Wh, a_src03 + l * DH, a_dst03 + l * DH, es, ed);
    fill_f32<<<(int)((nH + TB - 1) / TB), TB, 0, stream>>>(emax, nH, -INFINITY);
    fill_f32<<<(int)((nH + TB - 1) / TB), TB, 0, stream>>>(denom, nH, 0.f);
    fill_f32<<<gNC, TB, 0, stream>>>(agg, nNC, 0.f);
    edge_max_k<8><<<gE, TB, 0, stream>>>(src, dst, es, ed, emax);
    edge_den_k<8><<<gE, TB, 0, stream>>>(src, dst, es, ed, emax, denom);
    edge_agg_k<8, 4><<<gAgg, TB, 0, stream>>>(src, dst, es, ed, emax, denom, Wh, agg);
    gat_common(gamma03 + l * DH, beta03 + l * DH);
  }

  // 3) final single-head layer (W3 already [128,128] row-major)
  {
    gemm_n128_wmma<<<gGemm, TB, 0, stream>>>(h, W3, nullptr, Wh);
    long nH = (long)NN;
    attn_proj<1, 0><<<(int)((nH + TB - 1) / TB), TB, 0, stream>>>(Wh, a_src3, a_dst3, es, ed);
    fill_f32<<<(int)((nH + TB - 1) / TB), TB, 0, stream>>>(emax, nH, -INFINITY);
    fill_f32<<<(int)((nH + TB - 1) / TB), TB, 0, stream>>>(denom, nH, 0.f);
    fill_f32<<<gNC, TB, 0, stream>>>(agg, nNC, 0.f);
    edge_max_k<1><<<gE, TB, 0, stream>>>(src, dst, es, ed, emax);
    edge_den_k<1><<<gE, TB, 0, stream>>>(src, dst, es, ed, emax, denom);
    edge_agg_k<1, 7><<<gAgg, TB, 0, stream>>>(src, dst, es, ed, emax, denom, Wh, agg);
    gat_common(gamma3, beta3);
  }

  // 4) segment-mean pooling
  fill_f32<<<(GG * DH + TB - 1) / TB, TB, 0, stream>>>(pool, (long)GG * DH, 0.f);
  fill_f32<<<(GG + TB - 1) / TB, TB, 0, stream>>>(cnt, GG, 0.f);
  pool_k<<<gNC, TB, 0, stream>>>(h, indicator, pool, cnt);
  pool_div<<<(GG * DH + TB - 1) / TB, TB, 0, stream>>>(pool, cnt);

  // 5) MLP readout 128 -> 64 -> 32 -> 10
  mlp_layer<<<(GG * 64 + TB - 1) / TB, TB, 0, stream>>>(pool, mlp_W0, mlp_b0, t0, 128, 64, 1);
  mlp_layer<<<(GG * 32 + TB - 1) / TB, TB, 0, stream>>>(t0,   mlp_W1, mlp_b1, t1, 64, 32, 1);
  mlp_layer<<<(GG * 10 + TB - 1) / TB, TB, 0, stream>>>(t1,   mlp_W2, mlp_b2, out, 32, 10, 0);
}